// AdaptiveClassifier_83253646066172
// MI455X (gfx1250) — compile-verified
//
#include <hip/hip_runtime.h>
#include <hip/hip_bf16.h>

typedef __attribute__((ext_vector_type(2))) float v2f;
typedef __attribute__((ext_vector_type(8))) float v8f;

// ---------------------------------------------------------------------------
// Generic fp32 WMMA GEMM:  out[M,N] = opA(A)[M,K] @ B^T  (B stored row-major
// as [N,K] with leading dim ldb) + bias[N].
// One wave -> 32(M) x 64(N) output tile: per K-step, 2 A-loads + 4 B-loads
// feed 8 V_WMMA_F32_16X16X4_F32. A-row indices are clamped (not guarded) so
// the inner loop is branch-free. Loads are expressed as
// uniform_base[32-bit lane offset] so the compiler can use saddr-form
// global_load_b64 (SGPR base + VGPR offset + imm), keeping per-iteration
// address arithmetic scalar.
// ---------------------------------------------------------------------------
template<int RELU_A>
__global__ __launch_bounds__(32)
void wmma_gemm_f32_k(const float* __restrict__ A, const float* __restrict__ B,
                     const float* __restrict__ bias, float* __restrict__ out,
                     int M, int N, int K, int lda, int ldb, int ldo)
{
    const int lane = threadIdx.x & 31;
    const int nlo  = lane & 15;
    const int kb   = (lane >> 4) * 2;            // K pair {0,1} or {2,3}
    const int m0   = blockIdx.y * 32;
    const int n0   = blockIdx.x * 64;

    // Loop-invariant 32-bit per-lane element offsets (bases are uniform).
    const unsigned offA0 = (unsigned)(min(m0 + nlo,      M - 1) * lda + kb);
    const unsigned offA1 = (unsigned)(min(m0 + 16 + nlo, M - 1) * lda + kb);
    const unsigned offB0 = (unsigned)((n0 +  0 + nlo) * ldb + kb);
    const unsigned offB1 = (unsigned)((n0 + 16 + nlo) * ldb + kb);
    const unsigned offB2 = (unsigned)((n0 + 32 + nlo) * ldb + kb);
    const unsigned offB3 = (unsigned)((n0 + 48 + nlo) * ldb + kb);

    v8f a00 = {}, a01 = {}, a02 = {}, a03 = {};  // M-tile 0 x N-subtiles 0..3
    v8f a10 = {}, a11 = {}, a12 = {}, a13 = {};  // M-tile 1 x N-subtiles 0..3

    const float* Ak = A;   // uniform, advanced by scalar adds
    const float* Bk = B;
    for (int k0 = 0; k0 < K; k0 += 4) {
        v2f va0 = *(const v2f*)(Ak + offA0);
        v2f va1 = *(const v2f*)(Ak + offA1);
        v2f vb0 = *(const v2f*)(Bk + offB0);
        v2f vb1 = *(const v2f*)(Bk + offB1);
        v2f vb2 = *(const v2f*)(Bk + offB2);
        v2f vb3 = *(const v2f*)(Bk + offB3);
        if (RELU_A) {
            va0.x = fmaxf(va0.x, 0.f); va0.y = fmaxf(va0.y, 0.f);
            va1.x = fmaxf(va1.x, 0.f); va1.y = fmaxf(va1.y, 0.f);
        }

        a00 = __builtin_amdgcn_wmma_f32_16x16x4_f32(false, va0, false, vb0, (short)0, a00, false, false);
        a01 = __builtin_amdgcn_wmma_f32_16x16x4_f32(false, va0, false, vb1, (short)0, a01, false, false);
        a02 = __builtin_amdgcn_wmma_f32_16x16x4_f32(false, va0, false, vb2, (short)0, a02, false, false);
        a03 = __builtin_amdgcn_wmma_f32_16x16x4_f32(false, va0, false, vb3, (short)0, a03, false, false);
        a10 = __builtin_amdgcn_wmma_f32_16x16x4_f32(false, va1, false, vb0, (short)0, a10, false, false);
        a11 = __builtin_amdgcn_wmma_f32_16x16x4_f32(false, va1, false, vb1, (short)0, a11, false, false);
        a12 = __builtin_amdgcn_wmma_f32_16x16x4_f32(false, va1, false, vb2, (short)0, a12, false, false);
        a13 = __builtin_amdgcn_wmma_f32_16x16x4_f32(false, va1, false, vb3, (short)0, a13, false, false);

        Ak += 4;
        Bk += 4;
    }

    // D layout: VGPR r holds row (tile_m0 + r + 8*(lane>=16)), col n0+t*16+(lane&15)
    const int mhalf = (lane >> 4) * 8;
    v8f accs0[4] = {a00, a01, a02, a03};
    v8f accs1[4] = {a10, a11, a12, a13};
    #pragma unroll
    for (int t = 0; t < 4; ++t) {
        const int nc = n0 + t * 16 + nlo;
        const float bs = bias ? bias[nc] : 0.f;
        #pragma unroll
        for (int r = 0; r < 8; ++r) {
            const int mrA = m0 + mhalf + r;
            const int mrB = m0 + 16 + mhalf + r;
            if (mrA < M) out[(size_t)mrA * ldo + nc] = accs0[t][r] + bs;
            if (mrB < M) out[(size_t)mrB * ldo + nc] = accs1[t][r] + bs;
        }
    }
}

// ---------------------------------------------------------------------------
// seen[b,c] = sum_e relu(xh[b,e] + mh[c,e]) * p1[e] + p1_b   (c in [0,150))
// logits[b,150] = sum_e relu(xh[b,e]) * p1[e] + p1_b          ("unseen")
// One block per b; xh row + p1 staged in LDS; wave-per-c with shuffle reduce.
// ---------------------------------------------------------------------------
__global__ __launch_bounds__(256)
void seen_kernel(const float* __restrict__ xh, const float* __restrict__ mh,
                 const float* __restrict__ p1w, const float* __restrict__ p1b,
                 float* __restrict__ logits)
{
    const int b   = blockIdx.x;
    const int tid = threadIdx.x;
    __shared__ float sxh[1536];
    __shared__ float sp1[1536];
    for (int i = tid; i < 1536; i += 256) {
        sxh[i] = xh[(size_t)b * 1536 + i];
        sp1[i] = p1w[i];
    }
    __syncthreads();

    const int wave = tid >> 5;
    const int lane = tid & 31;
    const float bias = p1b[0];

    for (int c = wave; c < 151; c += 8) {
        float s = 0.f;
        if (c < 150) {
            const float* mrow = mh + (size_t)c * 1536;
            for (int e = lane; e < 1536; e += 32)
                s += fmaxf(sxh[e] + mrow[e], 0.f) * sp1[e];
        } else {
            for (int e = lane; e < 1536; e += 32)
                s += fmaxf(sxh[e], 0.f) * sp1[e];
        }
        #pragma unroll
        for (int off = 16; off > 0; off >>= 1)
            s += __shfl_xor(s, off, 32);
        if (lane == 0) logits[(size_t)b * 152 + c] = s + bias;
    }
}

// ---------------------------------------------------------------------------
// Masked classification loss for one row b; accumulates mean into accum[0].
// loss1: 2-way softmax over {logits[lab], logits[150]} (0 if lab==150)
// loss2: lse over all j except j==lab (when lab<150), target col 150.
// ---------------------------------------------------------------------------
__global__ __launch_bounds__(256)
void cls_loss_kernel(const float* __restrict__ logits, const int* __restrict__ lb,
                     float* __restrict__ accum)
{
    const int b   = blockIdx.x;
    const int tid = threadIdx.x;
    const int lab = lb[b];
    const float* row = logits + (size_t)b * 152;
    __shared__ float red[256];

    const bool incl = (tid < 151) && !(tid == lab && lab < 150);
    float v = incl ? row[tid] : -INFINITY;
    red[tid] = v;
    __syncthreads();
    for (int s = 128; s > 0; s >>= 1) {
        if (tid < s) red[tid] = fmaxf(red[tid], red[tid + s]);
        __syncthreads();
    }
    const float mx = red[0];
    __syncthreads();

    float e = incl ? expf(row[tid] - mx) : 0.f;
    red[tid] = e;
    __syncthreads();
    for (int s = 128; s > 0; s >>= 1) {
        if (tid < s) red[tid] += red[tid + s];
        __syncthreads();
    }

    if (tid == 0) {
        const float lse   = mx + logf(red[0]);
        const float l150  = row[150];
        const float loss2 = lse - l150;
        float loss1 = 0.f;
        if (lab < 150) {
            const float a  = row[lab];
            const float m2 = fmaxf(a, l150);
            loss1 = m2 + logf(expf(a - m2) + expf(l150 - m2)) - a;
        }
        atomicAdd(accum, (loss1 + loss2) * (1.0f / 1024.0f));
    }
}

// ---------------------------------------------------------------------------
// Row-normalize u (1024x64); also emit sq_n = ||u_norm||^2 per row.
// ---------------------------------------------------------------------------
__global__ __launch_bounds__(64)
void norm_kernel(const float* __restrict__ u, float* __restrict__ un,
                 float* __restrict__ sqn)
{
    const int b   = blockIdx.x;
    const int tid = threadIdx.x;
    __shared__ float red[64];
    const float v = u[(size_t)b * 64 + tid];
    red[tid] = v * v;
    __syncthreads();
    for (int s = 32; s > 0; s >>= 1) {
        if (tid < s) red[tid] += red[tid + s];
        __syncthreads();
    }
    const float nsq = red[0];
    const float inv = 1.0f / fmaxf(sqrtf(nsq), 1e-12f);
    un[(size_t)b * 64 + tid] = v * inv;
    if (tid == 0) sqn[b] = nsq * inv * inv;
}

// ---------------------------------------------------------------------------
// Pairwise margin losses from Gram matrix G = un @ un^T.
// accum[1]=pos_sum, [2]=pos_cnt, [3]=neg_sum, [4]=neg_cnt
// ---------------------------------------------------------------------------
__global__ __launch_bounds__(256)
void pair_loss_kernel(const float* __restrict__ G, const float* __restrict__ sqn,
                      const int* __restrict__ lb, float* __restrict__ accum)
{
    const int tid = threadIdx.x;
    const int idx = blockIdx.x * 256 + tid;
    const int i = idx >> 10;
    const int j = idx & 1023;

    const float sq   = sqn[i] + sqn[j] - 2.0f * G[idx];
    const float dist = (sq <= 0.f) ? 0.f : sqrtf(fmaxf(sq, 1e-16f));
    const bool same  = (lb[i] == lb[j]);
    const bool eye   = (i == j);
    const float pos  = (same && !eye) ? 1.f : 0.f;
    const float neg  = same ? 0.f : 1.f;
    const float pl   = fmaxf(dist - 0.7f, 0.f) * pos;
    const float nl   = fmaxf(1.4f - dist, 0.f) * neg;

    __shared__ float r0[256], r1[256], r2[256], r3[256];
    r0[tid] = pl; r1[tid] = pos; r2[tid] = nl; r3[tid] = neg;
    __syncthreads();
    for (int s = 128; s > 0; s >>= 1) {
        if (tid < s) {
            r0[tid] += r0[tid + s]; r1[tid] += r1[tid + s];
            r2[tid] += r2[tid + s]; r3[tid] += r3[tid + s];
        }
        __syncthreads();
    }
    if (tid == 0) {
        atomicAdd(accum + 1, r0[0]);
        atomicAdd(accum + 2, r1[0]);
        atomicAdd(accum + 3, r2[0]);
        atomicAdd(accum + 4, r3[0]);
    }
}

__global__ void init_kernel(float* accum)
{
    if (threadIdx.x < 8) accum[threadIdx.x] = 0.f;
}

__global__ void finalize_kernel(const float* __restrict__ accum, float* __restrict__ out)
{
    out[0] = accum[0]
           + accum[1] / fmaxf(accum[2], 1.f)
           + accum[3] / fmaxf(accum[4], 1.f);
}

// ---------------------------------------------------------------------------
extern "C" void kernel_launch(void* const* d_in, const int* in_sizes, int n_in,
                              void* d_out, int out_size, void* d_ws, size_t ws_size,
                              hipStream_t stream)
{
    (void)in_sizes; (void)n_in; (void)out_size; (void)ws_size;

    const float* x    = (const float*)d_in[0];
    const int*   lb   = (const int*)  d_in[1];
    const float* mem  = (const float*)d_in[2];
    const float* fc_w = (const float*)d_in[3];
    const float* fc_b = (const float*)d_in[4];
    const float* p1_w = (const float*)d_in[5];
    const float* p1_b = (const float*)d_in[6];
    const float* p2_w = (const float*)d_in[7];
    const float* p2_b = (const float*)d_in[8];
    float* out = (float*)d_out;

    // Workspace layout (floats): ~12.6 MB total
    float* ws     = (float*)d_ws;
    float* accum  = ws;                           // 8
    float* xh     = ws + 16;                      // 1024*1536
    float* mh     = xh + (size_t)1024 * 1536;     // 160*1536 (150 used)
    float* logits = mh + (size_t)160 * 1536;      // 1024*152 (151 used)
    float* u      = logits + (size_t)1024 * 152;  // 1024*64
    float* un     = u  + (size_t)1024 * 64;       // 1024*64
    float* sqn    = un + (size_t)1024 * 64;       // 1024
    float* G      = sqn + 1024;                   // 1024*1024

    init_kernel<<<1, 32, 0, stream>>>(accum);

    // xh = x @ Wx^T + fc_b          (Wx[e,d] = fc_w[e*1536 + d])
    wmma_gemm_f32_k<0><<<dim3(1536 / 64, 1024 / 32), 32, 0, stream>>>(
        x, fc_w, fc_b, xh, 1024, 1536, 768, 768, 1536, 1536);

    // mh = mem @ Wm^T               (Wm[e,d] = fc_w[e*1536 + 768 + d])
    wmma_gemm_f32_k<0><<<dim3(1536 / 64, 5), 32, 0, stream>>>(
        mem, fc_w + 768, nullptr, mh, 150, 1536, 768, 768, 1536, 1536);

    // logits[b, 0..150] (seen + unseen), fused relu reduction
    seen_kernel<<<1024, 256, 0, stream>>>(xh, mh, p1_w, p1_b, logits);

    // classification loss -> accum[0]
    cls_loss_kernel<<<1024, 256, 0, stream>>>(logits, lb, accum);

    // u = relu(xh) @ p2_w^T + p2_b
    wmma_gemm_f32_k<1><<<dim3(64 / 64, 1024 / 32), 32, 0, stream>>>(
        xh, p2_w, p2_b, u, 1024, 64, 1536, 1536, 1536, 64);

    // row-normalize, sq_n
    norm_kernel<<<1024, 64, 0, stream>>>(u, un, sqn);

    // G = un @ un^T
    wmma_gemm_f32_k<0><<<dim3(1024 / 64, 1024 / 32), 32, 0, stream>>>(
        un, un, nullptr, G, 1024, 1024, 64, 64, 64, 1024);

    // pairwise margin losses -> accum[1..4]
    pair_loss_kernel<<<4096, 256, 0, stream>>>(G, sqn, lb, accum);

    finalize_kernel<<<1, 1, 0, stream>>>(accum, out);
}